// SLoRA_78872779424006
// MI455X (gfx1250) — compile-verified
//
#include <hip/hip_runtime.h>

typedef __attribute__((ext_vector_type(2))) float v2f;
typedef __attribute__((ext_vector_type(4))) float v4f;
typedef __attribute__((ext_vector_type(8))) float v8f;

#define T_TOK 16384
#define D_IN  4096
#define O_OUT 4096
#define E_ADP 8
#define R_RANK 64

// ---------------- routing pass ----------------

__global__ void slora_init_counts(int* counts) {
    if (threadIdx.x < E_ADP) counts[threadIdx.x] = 0;
}

__global__ void slora_build_lists(const int* __restrict__ ids,
                                  int* __restrict__ counts,
                                  int* __restrict__ lists) {
    int t = blockIdx.x * blockDim.x + threadIdx.x;
    if (t < T_TOK) {
        int e = ids[t];
        int p = atomicAdd(&counts[e], 1);
        lists[e * T_TOK + p] = t;
    }
}

// ---------------- stage 1: H = gather(x) @ A_e  (16 tok x 64 rank per block) ----------------

__global__ __launch_bounds__(128)
void slora_stage1(const float* __restrict__ x, const float* __restrict__ A,
                  const int* __restrict__ counts, const int* __restrict__ lists,
                  float* __restrict__ H) {
    __shared__ float Xs[16][68];    // [token row][k]      (pad 68: banks 4m+c)
    __shared__ float As[64][68];    // transposed: [n][k]  (pad 68: banks 4n+c)
    __shared__ int   tokS[16];

    const int e     = blockIdx.y;
    const int cnt   = counts[e];
    const int tile0 = blockIdx.x * 16;
    if (tile0 >= cnt) return;                       // uniform exit, EXEC stays full

    const int tid  = threadIdx.x;
    const int lane = tid & 31;
    const int wave = tid >> 5;                      // 0..3 -> 16-wide slice of R
    const int ln   = lane & 15;
    const int h    = lane >> 4;                     // K-pair select (ISA A/B layout)

    if (tid < 16) {
        int p = tile0 + tid;
        if (p > cnt - 1) p = cnt - 1;               // clamp partial tile (idempotent dup)
        tokS[tid] = lists[e * T_TOK + p];
    }
    __syncthreads();

    const float* Ae = A + (size_t)e * D_IN * R_RANK;
    v8f acc = {0.f, 0.f, 0.f, 0.f, 0.f, 0.f, 0.f, 0.f};
    const int nloc = wave * 16 + ln;

    for (int k0 = 0; k0 < D_IN; k0 += 64) {
        // X tile: 16 x 64 = 256 x 128b, 2 per thread. x is streamed once -> TH=NT,
        // keeps A/B/H/lists resident in the 192MB L2.
        #pragma unroll
        for (int i = 0; i < 2; ++i) {
            int fid = tid + 128 * i;
            int row = fid >> 4;
            int c4  = (fid & 15) * 4;
            v4f v = __builtin_nontemporal_load(
                (const v4f*)(x + (size_t)tokS[row] * D_IN + k0 + c4));
            *(v4f*)&Xs[row][c4] = v;
        }
        // A tile: 64 x 64, transposed into As[n][k] (L2-resident, default RT)
        #pragma unroll
        for (int i = 0; i < 8; ++i) {
            int fid = tid + 128 * i;
            int kr  = fid >> 4;
            int n4  = (fid & 15) * 4;
            v4f v = *(const v4f*)(Ae + (size_t)(k0 + kr) * R_RANK + n4);
            As[n4 + 0][kr] = v.x;
            As[n4 + 1][kr] = v.y;
            As[n4 + 2][kr] = v.z;
            As[n4 + 3][kr] = v.w;
        }
        __syncthreads();

        #pragma unroll
        for (int kk = 0; kk < 64; kk += 4) {
            v2f a, b;
            a.x = Xs[ln][kk + 2 * h];
            a.y = Xs[ln][kk + 2 * h + 1];
            b.x = As[nloc][kk + 2 * h];
            b.y = As[nloc][kk + 2 * h + 1];
            acc = __builtin_amdgcn_wmma_f32_16x16x4_f32(false, a, false, b,
                                                        (short)0, acc, false, false);
        }
        __syncthreads();
    }

    #pragma unroll
    for (int r = 0; r < 8; ++r) {
        int m = r + 8 * h;
        H[(size_t)tokS[m] * R_RANK + nloc] = acc[r];
    }
}

// ---------------- stage 2: out = base + s_e * (H @ B_e)  (16 tok x 64 out per block) ----------------

__global__ __launch_bounds__(128)
void slora_stage2(const float* __restrict__ base, const float* __restrict__ B,
                  const float* __restrict__ scaling,
                  const int* __restrict__ counts, const int* __restrict__ lists,
                  const float* __restrict__ H, float* __restrict__ out) {
    __shared__ float Hs[16][68];    // [token row][r]
    __shared__ float Bs[64][68];    // transposed: [o_local][r]
    __shared__ int   tokS[16];

    const int e     = blockIdx.z;
    const int cnt   = counts[e];
    const int tile0 = blockIdx.x * 16;
    if (tile0 >= cnt) return;
    const int n0 = blockIdx.y * 64;

    const int tid  = threadIdx.x;
    const int lane = tid & 31;
    const int wave = tid >> 5;
    const int ln   = lane & 15;
    const int h    = lane >> 4;

    if (tid < 16) {
        int p = tile0 + tid;
        if (p > cnt - 1) p = cnt - 1;
        tokS[tid] = lists[e * T_TOK + p];
    }
    __syncthreads();

    // H tile: 16 x 64 (small, reused across 64 n-blocks -> default RT, L2 hit)
    #pragma unroll
    for (int i = 0; i < 2; ++i) {
        int fid = tid + 128 * i;
        int row = fid >> 4;
        int c4  = (fid & 15) * 4;
        v4f v = *(const v4f*)(H + (size_t)tokS[row] * R_RANK + c4);
        *(v4f*)&Hs[row][c4] = v;
    }
    // B tile: 64 (rank) x 64 (out cols), transposed into Bs[o][r] (L2-resident)
    const float* Be = B + (size_t)e * R_RANK * O_OUT;
    #pragma unroll
    for (int i = 0; i < 8; ++i) {
        int fid = tid + 128 * i;
        int kr  = fid >> 4;
        int n4  = (fid & 15) * 4;
        v4f v = *(const v4f*)(Be + (size_t)kr * O_OUT + n0 + n4);
        Bs[n4 + 0][kr] = v.x;
        Bs[n4 + 1][kr] = v.y;
        Bs[n4 + 2][kr] = v.z;
        Bs[n4 + 3][kr] = v.w;
    }
    __syncthreads();

    const int nloc = wave * 16 + ln;
    v8f acc = {0.f, 0.f, 0.f, 0.f, 0.f, 0.f, 0.f, 0.f};
    #pragma unroll
    for (int kk = 0; kk < 64; kk += 4) {
        v2f a, b;
        a.x = Hs[ln][kk + 2 * h];
        a.y = Hs[ln][kk + 2 * h + 1];
        b.x = Bs[nloc][kk + 2 * h];
        b.y = Bs[nloc][kk + 2 * h + 1];
        acc = __builtin_amdgcn_wmma_f32_16x16x4_f32(false, a, false, b,
                                                    (short)0, acc, false, false);
    }

    const float s = scaling[e];
    const int   n = n0 + nloc;
    #pragma unroll
    for (int r = 0; r < 8; ++r) {
        int m = r + 8 * h;
        size_t off = (size_t)tokS[m] * O_OUT + n;
        // base read once, out written once -> stream both past L2 with NT hints
        float b = __builtin_nontemporal_load(base + off);
        __builtin_nontemporal_store(b + s * acc[r], out + off);
    }
}

// ---------------- launch ----------------

extern "C" void kernel_launch(void* const* d_in, const int* in_sizes, int n_in,
                              void* d_out, int out_size, void* d_ws, size_t ws_size,
                              hipStream_t stream) {
    const float* x    = (const float*)d_in[0];
    const float* base = (const float*)d_in[1];
    const float* A    = (const float*)d_in[2];
    const float* Bm   = (const float*)d_in[3];
    const float* scal = (const float*)d_in[4];
    const int*   ids  = (const int*)d_in[5];
    float*       out  = (float*)d_out;

    char* ws     = (char*)d_ws;
    int*  counts = (int*)ws;                                     // E ints
    int*  lists  = (int*)(ws + 256);                             // E*T ints (512 KB)
    float* H     = (float*)(ws + 256 + (size_t)E_ADP * T_TOK * sizeof(int)); // T*R floats (4 MB)

    slora_init_counts<<<1, 32, 0, stream>>>(counts);
    slora_build_lists<<<T_TOK / 256, 256, 0, stream>>>(ids, counts, lists);
    slora_stage1<<<dim3(T_TOK / 16, E_ADP), 128, 0, stream>>>(x, A, counts, lists, H);
    slora_stage2<<<dim3(T_TOK / 16, O_OUT / 64, E_ADP), 128, 0, stream>>>(
        base, Bm, scal, counts, lists, H, out);
}